// fab_penalty_ls_curve_84885733638905
// MI455X (gfx1250) — compile-verified
//
#include <hip/hip_runtime.h>
#include <math.h>
#include <stdint.h>

// Problem geometry (fixed by the reference: eps is 4096x4096 f32, mirrored to 4096x8192)
#define H   4096
#define W   4096
#define W2  8192

// Tiling: 64x64 output tile per 256-thread (8-wave) block, halo of 2 on each side.
#define TI   64
#define TJ   64
#define HALO 2
#define LTI  (TI + 2 * HALO)   // 68
#define LTJ  (TJ + 2 * HALO)   // 68

typedef __attribute__((ext_vector_type(2))) float v2f;
typedef __attribute__((ext_vector_type(8))) float v8f;

// ---- CDNA5 async global->LDS copy path (ASYNCcnt), per CDNA5 ISA ----
// Per-lane gather: each lane copies one dword from its global address to its LDS byte address. Generic pointers to __shared__ carry the
// LDS byte offset in their low 32 bits (aperture encoding), which is exactly what VDST expects.
__device__ __forceinline__ void async_copy_b32(uint32_t lds_byte_addr, const void* gptr) {
  asm volatile("global_load_async_to_lds_b32 %0, %1, off"
               :: "v"(lds_byte_addr), "v"((uint64_t)(uintptr_t)gptr)
               : "memory");
}
__device__ __forceinline__ void wait_asynccnt0() {
  asm volatile("s_wait_asynccnt 0" ::: "memory");
}

__global__ void fab_zero_out(float* out) { out[0] = 0.0f; }

__global__ __launch_bounds__(256)
void fab_penalty_ls_curve_kernel(const float* __restrict__ eps,
                                 const float* __restrict__ gs,
                                 float* __restrict__ out) {
  __shared__ float tile[LTI * LTJ];
  __shared__ float wsum[8];

  const int tx = threadIdx.x;           // 0..63  -> column within tile (contiguous dim)
  const int ty = threadIdx.y;           // 0..3
  const int t  = ty * 64 + tx;
  const int bj = blockIdx.x * TJ;       // mirrored-space column origin
  const int bi = blockIdx.y * TI;       // row origin
  const int baseI = bi - HALO;
  const int baseJ = bj - HALO;

  // ---- Stage halo'd tile into LDS with async per-lane gathers (mirror+clamp folded in) ----
  const uint32_t lds_base = (uint32_t)(uintptr_t)(&tile[0]);
  for (int idx = t; idx < LTI * LTJ; idx += 256) {
    const int u  = idx / LTJ;
    const int v  = idx - u * LTJ;
    int gi = baseI + u;  gi = min(max(gi, 0), H - 1);     // clamped halo values are never
    int gj = baseJ + v;  gj = min(max(gj, 0), W2 - 1);    // read by the boundary formulas
    const int pj = (gj < W) ? gj : (2 * W - 1 - gj);      // mirror: col j -> eps[:, 8191-j]
    async_copy_b32(lds_base + (uint32_t)idx * 4u, eps + ((size_t)gi * W + pj));
  }
  wait_asynccnt0();
  __syncthreads();

  // ---- Fused derivative / curvature evaluation (all reads from LDS, branchless stencils) ----
  const float d      = gs[0];
  const float inv_d  = 1.0f / d;
  const float inv_2d = 0.5f * inv_d;
  const float SC     = 1e-12f;
  const float PI_D   = 2.855993321445267f;        // pi / (1.1 * MIN_FEATURE_SIZE)
  const float EVF    = 1.6666666666666667e-33f;   // 1e-32 / 6

  auto E = [&](int u, int v) -> float { return tile[u * LTJ + v]; };

  // Unified grad operator: forward/backward/central collapse to
  // (E[a+dp] - E[a-dm]) * s with dp=(a<N-1), dm=(a>0), s = (dp&&dm) ? 1/2d : 1/d.
  auto par = [&](int a, int n, int& dp, int& dm, float& s) {
    dp = (a < n - 1) ? 1 : 0;
    dm = (a > 0) ? 1 : 0;
    s  = (dp + dm == 2) ? inv_2d : inv_d;
  };

  float acc = 0.0f;
  const int j = bj + tx;          // this thread's mirrored column (fixed)
  const int v = tx + HALO;

  int dpj, dmj;  float sj;  par(j, W2, dpj, dmj, sj);               // column params (per lane)
  int dpjp, dmjp; float sjp; par(j + dpj, W2, dpjp, dmjp, sjp);     // params of column j+dpj
  int dpjm, dmjm; float sjm; par(j - dmj, W2, dpjm, dmjm, sjm);     // params of column j-dmj

  #pragma unroll 4
  for (int r = 0; r < 16; ++r) {
    const int i = bi + ty * 16 + r;     // wave-uniform row -> row params land in SALU
    const int u = ty * 16 + r + HALO;

    int dpi, dmi;  float si;  par(i, H, dpi, dmi, si);
    int dpip, dmip; float sip; par(i + dpi, H, dpip, dmip, sip);
    int dpim, dmim; float sim; par(i - dmi, H, dpim, dmim, sim);

    // first derivatives at the point
    const float ex = (E(u + dpi, v) - E(u - dmi, v)) * si + SC;
    const float ey = (E(u, v + dpj) - E(u, v - dmj)) * sj + SC;

    // eps_xx = d/dx of eps_x: eps_x evaluated at rows i+dpi and i-dmi
    const float ex_p = (E(u + dpi + dpip, v) - E(u + dpi - dmip, v)) * sip + SC;
    const float ex_m = (E(u - dmi + dpim, v) - E(u - dmi - dmim, v)) * sim + SC;
    const float exx  = (ex_p - ex_m) * si;

    // eps_xy = d/dy of eps_x: eps_x at columns j+dpj / j-dmj (row params unchanged)
    const float ex_jp = (E(u + dpi, v + dpj) - E(u - dmi, v + dpj)) * si + SC;
    const float ex_jm = (E(u + dpi, v - dmj) - E(u - dmi, v - dmj)) * si + SC;
    const float exy   = (ex_jp - ex_jm) * sj;

    // eps_yy = d/dy of eps_y: eps_y at columns j+dpj / j-dmj with their own params
    const float ey_p = (E(u, v + dpj + dpjp) - E(u, v + dpj - dmjp)) * sjp + SC;
    const float ey_m = (E(u, v - dmj + dpjm) - E(u, v - dmj - dmjm)) * sjm + SC;
    const float eyy  = (ey_p - ey_m) * sj;

    const float e0  = E(u, v);
    const float ev  = fmaxf(sqrtf(ex * ex + ey * ey), EVF);
    const float num = ex * ex * eyy - 2.0f * ex * ey * exy + ey * ey * exx;
    const float k   = num / (ev * ev * ev);
    const float curve = fabsf(k * atanf(ev / e0)) - PI_D;
    // nansum(maximum(curve,0)): NaN curve contributes 0; fmaxf(NaN,0)==0 matches exactly.
    acc += fmaxf(curve, 0.0f);
  }
  acc *= d * d;   // ALPHA * grid_size^2

  // ---- wave32 reduction via V_WMMA_F32_16X16X4_F32 (exact f32) ----
  // A (16x4, 2 VGPRs): lane L<16 holds row L, K={0,1}; lane L>=16 holds row L-16, K={2,3}.
  // With A[lane] = (acc, 0) and B = ones(4x16) (layout-independent), D[m,n] = p_m + p_{m+16}.
  // Summing a lane's 8 D regs gives rows 0-7 (lanes<16) or rows 8-15 (lanes>=16);
  // one shfl_xor(16) completes the wave sum. EXEC is all ones here (no divergence).
  float wave_total;
#if __has_builtin(__builtin_amdgcn_wmma_f32_16x16x4_f32)
  {
    v2f a; a[0] = acc; a[1] = 0.0f;
    v2f b; b[0] = 1.0f; b[1] = 1.0f;
    v8f c = {};
    v8f dm8 = __builtin_amdgcn_wmma_f32_16x16x4_f32(false, a, false, b,
                                                    (short)0, c, false, false);
    float s = ((dm8[0] + dm8[1]) + (dm8[2] + dm8[3])) +
              ((dm8[4] + dm8[5]) + (dm8[6] + dm8[7]));
    wave_total = s + __shfl_xor(s, 16, 32);
  }
#else
  {
    float s = acc;
    #pragma unroll
    for (int o = 16; o > 0; o >>= 1) s += __shfl_down(s, o, 32);
    wave_total = s;   // valid in lane 0
  }
#endif

  const int lane = t & 31, wid = t >> 5;
  if (lane == 0) wsum[wid] = wave_total;
  __syncthreads();
  if (t == 0) {
    float s = 0.0f;
    #pragma unroll
    for (int w = 0; w < 8; ++w) s += wsum[w];
    atomicAdd(out, s);
  }
}

extern "C" void kernel_launch(void* const* d_in, const int* in_sizes, int n_in,
                              void* d_out, int out_size, void* d_ws, size_t ws_size,
                              hipStream_t stream) {
  const float* eps = (const float*)d_in[0];   // 4096*4096 f32
  const float* gs  = (const float*)d_in[1];   // scalar grid_size
  float* out = (float*)d_out;                 // 1 f32 (harness poisons it -> zero first)

  fab_zero_out<<<1, 1, 0, stream>>>(out);

  dim3 block(64, 4);                 // 256 threads = 8 wave32
  dim3 grid(W2 / TJ, H / TI);        // 128 x 64 = 8192 blocks over the mirrored field
  fab_penalty_ls_curve_kernel<<<grid, block, 0, stream>>>(eps, gs, out);
}